// Cross_Fusion_1047972020964
// MI455X (gfx1250) — compile-verified
//
#include <hip/hip_runtime.h>
#include <hip/hip_bf16.h>

#define BB 16
#define LL 1024
#define DD 128
#define HH 256

typedef float v2f __attribute__((ext_vector_type(2)));
typedef float v8f __attribute__((ext_vector_type(8)));

__device__ __forceinline__ v8f wmma_f32(v2f a, v2f b, v8f c) {
    // D = A(16x4 f32) * B(4x16 f32) + C(16x16 f32)
    return __builtin_amdgcn_wmma_f32_16x16x4_f32(
        /*neg_a=*/false, a, /*neg_b=*/false, b,
        /*c_mod=*/(short)0, c, /*reuse_a=*/false, /*reuse_b=*/false);
}

// ---------------------------------------------------------------------------
// Kernel 1: c1[b,d] = sum_t y[b,t,d] / len_y[b]  ;  c2 likewise from z.
// ---------------------------------------------------------------------------
__global__ void reduce_kernel(const float* __restrict__ y, const float* __restrict__ z,
                              const int* __restrict__ len_y, const int* __restrict__ len_z,
                              float* __restrict__ C1, float* __restrict__ C2) {
    const int b = blockIdx.x;
    const int d = threadIdx.x;            // 128 threads
    const float* yb = y + (size_t)b * LL * DD;
    const float* zb = z + (size_t)b * LL * DD;
    float ay = 0.0f, az = 0.0f;
    for (int t = 0; t < LL; ++t) {
        ay += yb[(size_t)t * DD + d];
        az += zb[(size_t)t * DD + d];
    }
    C1[b * DD + d] = ay / (float)len_y[b];
    C2[b * DD + d] = az / (float)len_z[b];
}

// ---------------------------------------------------------------------------
// Kernel 2: one block per (mlp m, batch b). 256 threads.
//   h = gelu_exact(c @ w1 + b1) ;  out = h @ w2 + b2
// ---------------------------------------------------------------------------
struct MlpParams {
    const float* w1[4];
    const float* b1[4];
    const float* w2[4];
    const float* b2[4];
};

__global__ void mlp_kernel(MlpParams p, const float* __restrict__ C1,
                           const float* __restrict__ C2, float* __restrict__ OUT) {
    const int m = blockIdx.x;   // 0:W1(c1) 1:B1(c1) 2:W2(c2) 3:B2(c2)
    const int b = blockIdx.y;
    const int tid = threadIdx.x;  // 256
    __shared__ float cs[DD];
    __shared__ float hs[HH];
    const float* c = (m < 2) ? C1 : C2;
    if (tid < DD) cs[tid] = c[b * DD + tid];
    __syncthreads();
    {
        const float* w = p.w1[m];
        float acc = p.b1[m][tid];
        #pragma unroll 4
        for (int d = 0; d < DD; ++d) acc += cs[d] * w[d * HH + tid];
        // exact gelu: 0.5*x*(1+erf(x/sqrt(2)))
        hs[tid] = 0.5f * acc * (1.0f + erff(acc * 0.70710678118654752f));
    }
    __syncthreads();
    if (tid < DD) {
        const float* w = p.w2[m];
        float acc = p.b2[m][tid];
        #pragma unroll 4
        for (int j = 0; j < HH; ++j) acc += hs[j] * w[j * DD + tid];
        OUT[((size_t)m * BB + b) * DD + tid] = acc;
    }
}

// ---------------------------------------------------------------------------
// Kernel 3: s = 1 + (W1+W2)/2 ; bias = (b1+b2)/2
// ---------------------------------------------------------------------------
__global__ void combine_kernel(const float* __restrict__ OUT,
                               float* __restrict__ S, float* __restrict__ Bias) {
    const int b = blockIdx.x;
    const int d = threadIdx.x; // 128
    const float o0 = OUT[((size_t)0 * BB + b) * DD + d];
    const float o1 = OUT[((size_t)1 * BB + b) * DD + d];
    const float o2 = OUT[((size_t)2 * BB + b) * DD + d];
    const float o3 = OUT[((size_t)3 * BB + b) * DD + d];
    S[b * DD + d]    = 1.0f + 0.5f * (o0 + o2);
    Bias[b * DD + d] = 0.5f * (o1 + o3);
}

// ---------------------------------------------------------------------------
// Kernel 4: forward ragged DFT + filter fused epilogue.
// grid = (L/32, B), block = 128 (4 waves).
// Wave w: rows [kb, kb+16) with kb = bx*32 + (w>>1)*16, cols [dbase, dbase+64)
// with dbase = (w&1)*64  -> 4 d-tiles, 8 v8f accumulators (no spills).
// E[k,t] = cos(2pi r/l) - i sin(2pi r/l), r = (k*t) mod l, masked k<l,t<l.
// fr = xr*s + bias ; fi = xi*s   (xi = -sum sin * x)
// HW v_sin/v_cos take revolutions -> pass r/l directly.
// ---------------------------------------------------------------------------
__global__ void dft_fwd_kernel(const float* __restrict__ x, const int* __restrict__ len_x,
                               const float* __restrict__ S, const float* __restrict__ Bias,
                               float* __restrict__ FR, float* __restrict__ FI) {
    const int b = blockIdx.y;
    const int l = len_x[b];
    const float inv_l = 1.0f / (float)l;
    const int lane  = threadIdx.x & 31;
    const int w     = threadIdx.x >> 5;
    const int kb    = blockIdx.x * 32 + (w >> 1) * 16;
    const int dbase = (w & 1) * 64;
    const int mrow  = kb + (lane & 15);       // A-matrix row (M)
    const int c0    = (lane < 16) ? 0 : 2;    // K offset for this half-wave
    const int dn    = lane & 15;              // B-matrix column within tile
    const bool rowOK = (mrow < l);

    // incremental r = (mrow * col) % l for cols t+c0 and t+c0+1
    int ra = (mrow * c0) % l;
    int rb = (mrow * (c0 + 1)) % l;
    const int step = (4 * mrow) % l;

    const float* xb = x + (size_t)b * LL * DD;
    const int off0 = c0 * DD + dbase + dn;    // reg0 element offset at t=0

    v8f accR[4], accI[4];
    v8f zero = {0.f,0.f,0.f,0.f,0.f,0.f,0.f,0.f};
    #pragma unroll
    for (int j = 0; j < 4; ++j) { accR[j] = zero; accI[j] = zero; }

    // preload B operands for t = 0
    v2f Bcur[4];
    #pragma unroll
    for (int j = 0; j < 4; ++j) {
        Bcur[j].x = xb[off0 + j * 16];
        Bcur[j].y = xb[off0 + DD + j * 16];
    }

    #pragma unroll 1
    for (int t = 0; t < LL; t += 4) {
        // prefetch next K-step's B operands (software pipeline, depth 1)
        const int offn = (t + 4 < LL) ? (off0 + (t + 4) * DD) : off0;
        v2f Bnxt[4];
        #pragma unroll
        for (int j = 0; j < 4; ++j) {
            Bnxt[j].x = xb[offn + j * 16];
            Bnxt[j].y = xb[offn + DD + j * 16];
        }

        const float fa = (float)ra * inv_l;
        const float fb = (float)rb * inv_l;
        float ca = __builtin_amdgcn_cosf(fa);
        float sa = -__builtin_amdgcn_sinf(fa);
        float cb = __builtin_amdgcn_cosf(fb);
        float sb = -__builtin_amdgcn_sinf(fb);
        const bool okA = rowOK & ((t + c0) < l);
        const bool okB = rowOK & ((t + c0 + 1) < l);
        v2f Acos, Asin;
        Acos.x = okA ? ca : 0.0f;  Acos.y = okB ? cb : 0.0f;
        Asin.x = okA ? sa : 0.0f;  Asin.y = okB ? sb : 0.0f;
        ra += step; if (ra >= l) ra -= l;
        rb += step; if (rb >= l) rb -= l;

        #pragma unroll
        for (int j = 0; j < 4; ++j) {
            accR[j] = wmma_f32(Acos, Bcur[j], accR[j]);
            accI[j] = wmma_f32(Asin, Bcur[j], accI[j]);
        }
        #pragma unroll
        for (int j = 0; j < 4; ++j) Bcur[j] = Bnxt[j];
    }

    // Epilogue: apply per-(b,d) scale + bias, store fr/fi.
    const int rowAdd = (lane < 16) ? 0 : 8;
    #pragma unroll
    for (int j = 0; j < 4; ++j) {
        const int d = dbase + j * 16 + dn;
        const float sv = S[b * DD + d];
        const float bv = Bias[b * DD + d];
        #pragma unroll
        for (int r2 = 0; r2 < 8; ++r2) {
            const int krow = kb + r2 + rowAdd;
            const size_t idx = ((size_t)b * LL + krow) * DD + d;
            FR[idx] = accR[j][r2] * sv + bv;
            FI[idx] = accI[j][r2] * sv;
        }
    }
}

// ---------------------------------------------------------------------------
// Kernel 5: inverse ragged DFT, real part only.
// out[k,d] = (1/l) * sum_{t<l} ( cos(2pi r/l)*fr - sin(2pi r/l)*fi ), k<l else 0.
// Two WMMAs per tile accumulate into one accumulator (A_cos, A_negsin).
// Same wave tiling as forward: 16(k) x 64(d) per wave.
// ---------------------------------------------------------------------------
__global__ void dft_inv_kernel(const float* __restrict__ FR, const float* __restrict__ FI,
                               const int* __restrict__ len_x, float* __restrict__ out) {
    const int b = blockIdx.y;
    const int l = len_x[b];
    const float inv_l = 1.0f / (float)l;
    const int lane  = threadIdx.x & 31;
    const int w     = threadIdx.x >> 5;
    const int kb    = blockIdx.x * 32 + (w >> 1) * 16;
    const int dbase = (w & 1) * 64;
    const int mrow  = kb + (lane & 15);
    const int c0    = (lane < 16) ? 0 : 2;
    const int dn    = lane & 15;
    const bool rowOK = (mrow < l);

    int ra = (mrow * c0) % l;
    int rb = (mrow * (c0 + 1)) % l;
    const int step = (4 * mrow) % l;

    const float* frb = FR + (size_t)b * LL * DD;
    const float* fib = FI + (size_t)b * LL * DD;
    const int off0 = c0 * DD + dbase + dn;

    v8f acc[4];
    v8f zero = {0.f,0.f,0.f,0.f,0.f,0.f,0.f,0.f};
    #pragma unroll
    for (int j = 0; j < 4; ++j) acc[j] = zero;

    // preload B operands for t = 0
    v2f BrC[4], BiC[4];
    #pragma unroll
    for (int j = 0; j < 4; ++j) {
        BrC[j].x = frb[off0 + j * 16];
        BrC[j].y = frb[off0 + DD + j * 16];
        BiC[j].x = fib[off0 + j * 16];
        BiC[j].y = fib[off0 + DD + j * 16];
    }

    #pragma unroll 1
    for (int t = 0; t < LL; t += 4) {
        const int offn = (t + 4 < LL) ? (off0 + (t + 4) * DD) : off0;
        v2f BrN[4], BiN[4];
        #pragma unroll
        for (int j = 0; j < 4; ++j) {
            BrN[j].x = frb[offn + j * 16];
            BrN[j].y = frb[offn + DD + j * 16];
            BiN[j].x = fib[offn + j * 16];
            BiN[j].y = fib[offn + DD + j * 16];
        }

        const float fa = (float)ra * inv_l;
        const float fb = (float)rb * inv_l;
        float ca = __builtin_amdgcn_cosf(fa);
        float sa = -__builtin_amdgcn_sinf(fa);   // coefficient of fi is -sin
        float cb = __builtin_amdgcn_cosf(fb);
        float sb = -__builtin_amdgcn_sinf(fb);
        const bool okA = rowOK & ((t + c0) < l);
        const bool okB = rowOK & ((t + c0 + 1) < l);
        v2f Acos, Asin;
        Acos.x = okA ? ca : 0.0f;  Acos.y = okB ? cb : 0.0f;
        Asin.x = okA ? sa : 0.0f;  Asin.y = okB ? sb : 0.0f;
        ra += step; if (ra >= l) ra -= l;
        rb += step; if (rb >= l) rb -= l;

        #pragma unroll
        for (int j = 0; j < 4; ++j) {
            acc[j] = wmma_f32(Acos, BrC[j], acc[j]);
            acc[j] = wmma_f32(Asin, BiC[j], acc[j]);
        }
        #pragma unroll
        for (int j = 0; j < 4; ++j) { BrC[j] = BrN[j]; BiC[j] = BiN[j]; }
    }

    const int rowAdd = (lane < 16) ? 0 : 8;
    #pragma unroll
    for (int j = 0; j < 4; ++j) {
        const int d = dbase + j * 16 + dn;
        #pragma unroll
        for (int r2 = 0; r2 < 8; ++r2) {
            const int krow = kb + r2 + rowAdd;
            const size_t idx = ((size_t)b * LL + krow) * DD + d;
            out[idx] = acc[j][r2] * inv_l;   // rows k>=l accumulated exactly 0
        }
    }
}

// ---------------------------------------------------------------------------
extern "C" void kernel_launch(void* const* d_in, const int* in_sizes, int n_in,
                              void* d_out, int out_size, void* d_ws, size_t ws_size,
                              hipStream_t stream) {
    const float* x  = (const float*)d_in[0];
    const float* y  = (const float*)d_in[1];
    const float* z  = (const float*)d_in[2];
    const int* len_x = (const int*)d_in[3];
    const int* len_y = (const int*)d_in[4];
    const int* len_z = (const int*)d_in[5];

    MlpParams p;
    // order: W1(6..9), B1(10..13), W2(14..17), B2(18..21); each: l1_w,l1_b,l2_w,l2_b
    for (int m = 0; m < 4; ++m) {
        p.w1[m] = (const float*)d_in[6 + 4 * m + 0];
        p.b1[m] = (const float*)d_in[6 + 4 * m + 1];
        p.w2[m] = (const float*)d_in[6 + 4 * m + 2];
        p.b2[m] = (const float*)d_in[6 + 4 * m + 3];
    }

    float* ws = (float*)d_ws;
    float* FR   = ws;                              // B*L*D
    float* FI   = FR + (size_t)BB * LL * DD;       // B*L*D
    float* C1v  = FI + (size_t)BB * LL * DD;       // B*D
    float* C2v  = C1v + BB * DD;                   // B*D
    float* MOUT = C2v + BB * DD;                   // 4*B*D
    float* Sv   = MOUT + 4 * BB * DD;              // B*D
    float* Bv   = Sv + BB * DD;                    // B*D

    reduce_kernel<<<dim3(BB), dim3(DD), 0, stream>>>(y, z, len_y, len_z, C1v, C2v);
    mlp_kernel<<<dim3(4, BB), dim3(HH), 0, stream>>>(p, C1v, C2v, MOUT);
    combine_kernel<<<dim3(BB), dim3(DD), 0, stream>>>(MOUT, Sv, Bv);
    dft_fwd_kernel<<<dim3(LL / 32, BB), dim3(128), 0, stream>>>(x, len_x, Sv, Bv, FR, FI);
    dft_inv_kernel<<<dim3(LL / 32, BB), dim3(128), 0, stream>>>(FR, FI, len_x, (float*)d_out);
}